// CartesianToJacobi_5437428596843
// MI455X (gfx1250) — compile-verified
//
#include <hip/hip_runtime.h>
#include <stdint.h>

// MI455X / gfx1250. Memory-bound problem (~109MB @ 23.3TB/s ≈ 4.7us floor).
// Strategy: async global->LDS staging (ASYNCcnt path), per-thread prefix-sum
// compute in f32 (matches reference numerics), coalesced b128 stores.

#define MB   128   // batches per block == threads per block
#define MSTR 136   // m-plane stride in float4 (pads async LDS writes to 2-phase)
#define QSTR 133   // q/v-plane stride in float4

__global__ __launch_bounds__(MB) void jacobi16_kernel(
    const float* __restrict__ mg, const float* __restrict__ qg,
    const float* __restrict__ vg, float* __restrict__ outq,
    float* __restrict__ outv)
{
    __shared__ float4 sm_m[4  * MSTR];   // m  tile: 4 chunks/batch
    __shared__ float4 sm_q[12 * QSTR];   // q  tile: 12 chunks/batch
    __shared__ float4 sm_v[12 * QSTR];   // v  tile: 12 chunks/batch

    const uint32_t tid = threadIdx.x;
    const size_t   bb  = (size_t)blockIdx.x * MB;

    const float4* gm = (const float4*)(mg + bb * 16);
    const float4* gq = (const float4*)(qg + bb * 48);
    const float4* gv = (const float4*)(vg + bb * 48);

    // ---- Stage block tile into LDS with gfx1250 async copies.
    // Global side: lane-consecutive 16B chunks (perfectly coalesced).
    // LDS side: chunk-transposed [chunk][batch] so compute reads are
    // lane-consecutive b128 (bank-conflict-free).
#pragma unroll
    for (int k = 0; k < 4; ++k) {
        uint32_t g = tid + k * MB;            // m chunk id 0..511
        uint32_t t = g >> 2, c = g & 3u;      // batch-in-tile, chunk-in-batch
        uint32_t dst = (uint32_t)(uintptr_t)&sm_m[c * MSTR + t];
        asm volatile("global_load_async_to_lds_b128 %0, %1, off"
                     :: "v"(dst), "v"(gm + g) : "memory");
    }
#pragma unroll
    for (int k = 0; k < 12; ++k) {
        uint32_t g = tid + k * MB;            // q/v chunk id 0..1535
        uint32_t t = g / 12u, c = g % 12u;
        uint32_t dq = (uint32_t)(uintptr_t)&sm_q[c * QSTR + t];
        uint32_t dv = (uint32_t)(uintptr_t)&sm_v[c * QSTR + t];
        asm volatile("global_load_async_to_lds_b128 %0, %1, off"
                     :: "v"(dq), "v"(gq + g) : "memory");
        asm volatile("global_load_async_to_lds_b128 %0, %1, off"
                     :: "v"(dv), "v"(gv + g) : "memory");
    }
    asm volatile("s_wait_asynccnt 0" ::: "memory");
    __syncthreads();

    // ---- Per-thread Jacobi transform (one batch per thread, all f32) ----
    float mm[16];
#pragma unroll
    for (int c = 0; c < 4; ++c) {
        float4 ch = sm_m[c * MSTR + tid];
        mm[4*c+0] = ch.x; mm[4*c+1] = ch.y; mm[4*c+2] = ch.z; mm[4*c+3] = ch.w;
    }
    float M = mm[0];
    float invP[16];                 // invP[i] = 1 / M[i-1], i >= 1
#pragma unroll
    for (int i = 1; i < 16; ++i) { invP[i] = 1.0f / M; M += mm[i]; }
    const float invT = 1.0f / M;    // 1 / M_tot

    // q pass then v pass (two passes keeps VGPR pressure low)
#pragma unroll
    for (int pass = 0; pass < 2; ++pass) {
        const float4* smx = pass ? sm_v : sm_q;
        float*        out = pass ? outv : outq;

        float xf[48];
#pragma unroll
        for (int c = 0; c < 12; ++c) {
            float4 ch = smx[c * QSTR + tid];
            xf[4*c+0] = ch.x; xf[4*c+1] = ch.y; xf[4*c+2] = ch.z; xf[4*c+3] = ch.w;
        }

        float o[48];
        float a0 = 0.f, a1 = 0.f, a2 = 0.f;   // prefix sums of m[j]*x[j][d]
#pragma unroll
        for (int i = 0; i < 16; ++i) {
            float x0 = xf[3*i], x1 = xf[3*i+1], x2 = xf[3*i+2];
            if (i > 0) {
                float ip = invP[i];
                o[3*i+0] = fmaf(-a0, ip, x0);   // x[i] - prefix/M[i-1]
                o[3*i+1] = fmaf(-a1, ip, x1);
                o[3*i+2] = fmaf(-a2, ip, x2);
            }
            a0 = fmaf(mm[i], x0, a0);
            a1 = fmaf(mm[i], x1, a1);
            a2 = fmaf(mm[i], x2, a2);
        }
        o[0] = a0 * invT; o[1] = a1 * invT; o[2] = a2 * invT;  // row 0

        float4* og = (float4*)(out + (bb + tid) * 48);         // 16B aligned
#pragma unroll
        for (int k = 0; k < 12; ++k) {
            float4 st;
            st.x = o[4*k+0]; st.y = o[4*k+1]; st.z = o[4*k+2]; st.w = o[4*k+3];
            og[k] = st;
        }
    }
}

extern "C" void kernel_launch(void* const* d_in, const int* in_sizes, int n_in,
                              void* d_out, int out_size, void* d_ws, size_t ws_size,
                              hipStream_t stream) {
    const float* m = (const float*)d_in[0];   // (B,16)
    const float* q = (const float*)d_in[1];   // (B,16,3)
    const float* v = (const float*)d_in[2];   // (B,16,3)
    const int B = in_sizes[0] / 16;           // 131072

    float* outq = (float*)d_out;              // (B,16,3)
    float* outv = outq + (size_t)B * 48;      // (B,16,3)

    const int grid = B / MB;                  // B is a multiple of 128
    jacobi16_kernel<<<grid, MB, 0, stream>>>(m, q, v, outq, outv);
}